// VQ_3006477107443
// MI455X (gfx1250) — compile-verified
//
#include <hip/hip_runtime.h>

// Problem constants (from reference)
#define B_  32
#define D_  256
#define H_  32
#define W_  32
#define K_  1024
#define HW  (H_ * W_)               // 1024
#define N_  (B_ * H_ * W_)          // 32768 tokens
#define ZQ_SIZE (B_ * D_ * H_ * W_) // 8388608
#define IDX_OFF  ZQ_SIZE
#define LOSS_OFF (ZQ_SIZE + N_)
#define BETA 0.25f

// Tiling for the distance GEMM
#define BM   64            // tokens per workgroup (4 row-tiles of 16)
#define NRT  (BM / 16)     // 4 row tiles
#define NDG  (D_ / 4)      // 64 k-groups of 4 (WMMA K=4)

typedef __attribute__((ext_vector_type(2))) float v2f;
typedef __attribute__((ext_vector_type(8))) float v8f;

// ---------------------------------------------------------------------------
// Kernel 1: codebook row norms ||c_k||^2 ; also zero the loss accumulator.
// One wave per codebook row (8 rows per 256-thread block).
// ---------------------------------------------------------------------------
__global__ __launch_bounds__(256)
void vq_norms_kernel(const float* __restrict__ cb,
                     float* __restrict__ cnorm,
                     float* __restrict__ loss_slot) {
    const int lane = threadIdx.x & 31;
    const int wave = threadIdx.x >> 5;
    const int row  = blockIdx.x * 8 + wave;
    const float* r = cb + (size_t)row * D_;
    float s = 0.f;
#pragma unroll
    for (int j = 0; j < D_ / 32; ++j) {
        float v = r[lane + 32 * j];
        s += v * v;
    }
#pragma unroll
    for (int m = 16; m >= 1; m >>= 1) s += __shfl_xor(s, m, 32);
    if (lane == 0) cnorm[row] = s;
    if (blockIdx.x == 0 && threadIdx.x == 0) *loss_slot = 0.f;
}

// ---------------------------------------------------------------------------
// Kernel 2: fused -2*z.c + ||c||^2 GEMM (fp32 WMMA 16x16x4) + argmin over K.
// Block = 256 threads = 8 waves, handles BM=64 tokens x all 1024 codes.
// Wave w owns columns [w*128, w*128+128) as 8 tiles of 16.
// A tile is staged in LDS *pre-swizzled into WMMA fragment layout*:
//   entry (dg, r, lane) = v2f{ A[4dg+2*(lane>>4)][r*16+(lane&15)],
//                              A[4dg+2*(lane>>4)+1][r*16+(lane&15)] }
// so the inner loop is one contiguous ds_load_b64 per fragment (bank-free).
// ---------------------------------------------------------------------------
__global__ __launch_bounds__(256)
void vq_argmin_kernel(const float* __restrict__ z_e,
                      const float* __restrict__ cb,
                      const float* __restrict__ cnorm,
                      int* __restrict__ ws_idx,
                      float* __restrict__ out_idx_f) {
    // 64KB: NDG*NRT*32 v2f entries. Reduction scratch aliased after barrier.
    __shared__ __align__(16) unsigned char smem[NDG * NRT * 32 * sizeof(v2f)];
    v2f*   sFrag = (v2f*)smem;
    float* sredv = (float*)smem;                       // BM*8 floats
    int*   sredk = (int*)(smem + BM * 8 * sizeof(float)); // BM*8 ints

    const int tid  = threadIdx.x;
    const int lane = tid & 31;
    const int wave = tid >> 5;
    const int half = lane >> 4;           // 0: lanes 0-15, 1: lanes 16-31
    const int l15  = lane & 15;

    const int n0  = blockIdx.x * BM;      // 64 tokens = two (b,h) rows
    const int b   = n0 >> 10;
    const int hw0 = n0 & (HW - 1);
    const float* gbase = z_e + (size_t)b * D_ * HW + hw0;

    // ---- stage A tile directly into fragment layout (8192 v2f, 32/thread)
#pragma unroll 4
    for (int t = 0; t < 32; ++t) {
        const int fidx = t * 256 + tid;   // (dg*NRT + r)*32 + flane
        const int fl   = fidx & 31;
        const int rdg  = fidx >> 5;
        const int r    = rdg & (NRT - 1);
        const int dg   = rdg >> 2;
        const int m    = r * 16 + (fl & 15);
        const int d0   = dg * 4 + 2 * (fl >> 4);
        v2f v;
        v.x = gbase[(size_t)d0 * HW + m];
        v.y = gbase[(size_t)(d0 + 1) * HW + m];
        sFrag[fidx] = v;
    }
    __syncthreads();

    float bestv[NRT][8];
    int   bestk[NRT][8];
#pragma unroll
    for (int r = 0; r < NRT; ++r)
#pragma unroll
        for (int i = 0; i < 8; ++i) { bestv[r][i] = 3.4e38f; bestk[r][i] = 0; }

    const int kwave = wave * 128;

    for (int kt = 0; kt < 8; ++kt) {
        const int k0 = kwave + kt * 16;

        v8f acc[NRT];
#pragma unroll
        for (int r = 0; r < NRT; ++r)
            acc[r] = (v8f){0.f, 0.f, 0.f, 0.f, 0.f, 0.f, 0.f, 0.f};

        // B fragment base: lane l -> codebook row k0+l15, dims d+2*half..+1
        const float* brow = cb + (size_t)(k0 + l15) * D_ + half * 2;

#pragma unroll 4
        for (int dg = 0; dg < NDG; ++dg) {
            const v2f bf = *(const v2f*)(brow + dg * 4); // 8B global (L2 hit)
            const v2f* afp = sFrag + (dg * NRT) * 32 + lane;
#pragma unroll
            for (int r = 0; r < NRT; ++r) {
                const v2f af = afp[r * 32];              // ds_load_b64
                acc[r] = __builtin_amdgcn_wmma_f32_16x16x4_f32(
                    /*neg_a=*/false, af, /*neg_b=*/false, bf,
                    /*c_mod=*/(short)0, acc[r],
                    /*reuse_a=*/false, /*reuse_b=*/false);
            }
        }

        // epilogue: d2 = ||c||^2 - 2*dot  (||z||^2 constant per row -> dropped)
        const float cn = cnorm[k0 + l15];
#pragma unroll
        for (int r = 0; r < NRT; ++r)
#pragma unroll
            for (int i = 0; i < 8; ++i) {
                const float v = cn - 2.0f * acc[r][i];
                if (v < bestv[r][i]) { bestv[r][i] = v; bestk[r][i] = k0 + l15; }
            }
    }

    // ---- reduce argmin across the 16 lanes of each half (width-16 butterfly)
#pragma unroll
    for (int r = 0; r < NRT; ++r)
#pragma unroll
        for (int i = 0; i < 8; ++i) {
            float v = bestv[r][i];
            int   k = bestk[r][i];
#pragma unroll
            for (int m = 8; m >= 1; m >>= 1) {
                const float ov = __shfl_xor(v, m, 16);
                const int   ok = __shfl_xor(k, m, 16);
                if (ov < v || (ov == v && ok < k)) { v = ov; k = ok; }
            }
            bestv[r][i] = v;
            bestk[r][i] = k;
        }

    // sFrag no longer needed by anyone after this barrier -> alias as scratch
    __syncthreads();

#pragma unroll
    for (int r = 0; r < NRT; ++r)
#pragma unroll
        for (int i = 0; i < 8; ++i) {
            if (l15 == 0) {
                const int mloc = r * 16 + i + half * 8;   // token within block
                sredv[mloc * 8 + wave] = bestv[r][i];
                sredk[mloc * 8 + wave] = bestk[r][i];
            }
        }
    __syncthreads();

    // ---- cross-wave reduce (8 column ranges), emit final index
    if (tid < BM) {
        float v = sredv[tid * 8];
        int   k = sredk[tid * 8];
#pragma unroll
        for (int w2 = 1; w2 < 8; ++w2) {
            const float ov = sredv[tid * 8 + w2];
            const int   ok = sredk[tid * 8 + w2];
            if (ov < v || (ov == v && ok < k)) { v = ov; k = ok; }
        }
        ws_idx[n0 + tid]    = k;
        out_idx_f[n0 + tid] = (float)k;
    }
}

// ---------------------------------------------------------------------------
// Kernel 3: z_q[b,d,h,w] = codebook[idx[n]][d] (NCHW), fused loss reduction.
// One block per (b,h): 32 tokens, coalesced stores along w.
// ---------------------------------------------------------------------------
__global__ __launch_bounds__(256)
void vq_gather_kernel(const float* __restrict__ z_e,
                      const float* __restrict__ cb,
                      const int* __restrict__ ws_idx,
                      float* __restrict__ zq,
                      float* __restrict__ loss_slot) {
    __shared__ int   s_idx[W_];
    __shared__ float s_wsum[8];

    const int tid = threadIdx.x;
    const int bh  = blockIdx.x;           // 0..1023
    const int b   = bh >> 5;
    const int h   = bh & 31;
    const int nbase = b * HW + h * W_;

    if (tid < W_) s_idx[tid] = ws_idx[nbase + tid];
    __syncthreads();

    const int w    = tid & 31;
    const int dgrp = tid >> 5;            // 0..7 -> 32 dims each
    const size_t gb = (size_t)b * D_ * HW + (size_t)h * W_ + w;
    const float* crow = cb + (size_t)s_idx[w] * D_;

    float local = 0.f;
#pragma unroll 4
    for (int dd = 0; dd < 32; ++dd) {
        const int d = dgrp * 32 + dd;
        const float c = crow[d];
        const float z = z_e[gb + (size_t)d * HW];
        zq[gb + (size_t)d * HW] = c;
        const float diff = c - z;
        local += diff * diff;
    }
#pragma unroll
    for (int m = 16; m >= 1; m >>= 1) local += __shfl_xor(local, m, 32);
    if ((tid & 31) == 0) s_wsum[tid >> 5] = local;
    __syncthreads();
    if (tid == 0) {
        float s = 0.f;
#pragma unroll
        for (int i = 0; i < 8; ++i) s += s_wsum[i];
        // loss_codebook + BETA*loss_commit = (1+BETA) * mean(sq diff)
        atomicAdd(loss_slot, s * ((1.0f + BETA) / (float)ZQ_SIZE));
    }
}

// ---------------------------------------------------------------------------
extern "C" void kernel_launch(void* const* d_in, const int* in_sizes, int n_in,
                              void* d_out, int out_size, void* d_ws, size_t ws_size,
                              hipStream_t stream) {
    const float* z_e = (const float*)d_in[0];   // [32,256,32,32] fp32
    const float* cb  = (const float*)d_in[1];   // [1024,256] fp32

    float* out        = (float*)d_out;
    float* out_idx_f  = out + IDX_OFF;          // indices as float
    float* loss_slot  = out + LOSS_OFF;         // single float

    float* cnorm  = (float*)d_ws;               // 1024 floats
    int*   ws_idx = (int*)d_ws + K_;            // 32768 ints

    vq_norms_kernel<<<K_ / 8, 256, 0, stream>>>(cb, cnorm, loss_slot);
    vq_argmin_kernel<<<N_ / BM, 256, 0, stream>>>(z_e, cb, cnorm, ws_idx, out_idx_f);
    vq_gather_kernel<<<B_ * H_, 256, 0, stream>>>(z_e, cb, ws_idx, out, loss_slot);
}